// VanillaGRU_65300682769024
// MI455X (gfx1250) — compile-verified
//
#include <hip/hip_runtime.h>
#include <hip/hip_bf16.h>
#include <math.h>

// ---------------------------------------------------------------------------
// VanillaGRU: BatchNorm1d(T) -> ReLU -> GRU(h=1) over T=80 -> ReLU -> FC(T->1)
// B=16384, T=80, F=24.  Memory-bound (~160 MB traffic vs ~0.4 GFLOP): stream
// x twice (stats pass + fused BN/WMMA gate projection), fp32 throughout.
// Gate projection uses V_WMMA_F32_16X16X4_F32.
// ---------------------------------------------------------------------------

#define T_DIM 80
#define F_DIM 24

typedef __attribute__((ext_vector_type(2))) float v2f;
typedef __attribute__((ext_vector_type(8))) float v8f;

// ---------------------------------------------------------------------------
// Kernel 1: per-t partial sums of x and x^2 over (B,F).  Each thread owns
// whole 24-float rows (6 aligned float4 loads, no div/mod).  Deterministic
// two-level reduction (no float atomics).  part[(t*nchunk+c)*2 + {0,1}].
// ---------------------------------------------------------------------------
__global__ void stats_partial_kernel(const float* __restrict__ x,
                                     float* __restrict__ part,
                                     int Bn, int nchunk, int bPerChunk) {
    const int t     = blockIdx.x;           // 0..T-1
    const int chunk = blockIdx.y;           // 0..nchunk-1
    const int tid   = threadIdx.x;

    float s = 0.f, s2 = 0.f;
    const int bEnd = min(Bn, (chunk + 1) * bPerChunk);
    for (int b = chunk * bPerChunk + tid; b < bEnd; b += blockDim.x) {
        const float* row = x + ((long)b * T_DIM + t) * F_DIM;  // 96B multiple
#pragma unroll
        for (int f = 0; f < F_DIM; f += 4) {
            float4 v = *(const float4*)(row + f);              // 16B aligned
            s  += v.x + v.y + v.z + v.w;
            s2 += v.x * v.x + v.y * v.y + v.z * v.z + v.w * v.w;
        }
    }
    __shared__ float sm[256];
    __shared__ float sm2[256];
    sm[tid] = s; sm2[tid] = s2;
    __syncthreads();
    for (int off = 128; off > 0; off >>= 1) {
        if (tid < off) { sm[tid] += sm[tid + off]; sm2[tid] += sm2[tid + off]; }
        __syncthreads();
    }
    if (tid == 0) {
        part[(t * nchunk + chunk) * 2 + 0] = sm[0];
        part[(t * nchunk + chunk) * 2 + 1] = sm2[0];
    }
}

// ---------------------------------------------------------------------------
// Kernel 2: finalize stats -> per-t affine (scale, bias) with gamma/beta/eps.
// ---------------------------------------------------------------------------
__global__ void stats_final_kernel(const float* __restrict__ part,
                                   const float* __restrict__ gamma,
                                   const float* __restrict__ beta,
                                   float* __restrict__ scalebias,
                                   int nchunk, float invN) {
    int t = threadIdx.x;
    if (t < T_DIM) {
        float s = 0.f, s2 = 0.f;
        for (int c = 0; c < nchunk; ++c) {
            s  += part[(t * nchunk + c) * 2 + 0];
            s2 += part[(t * nchunk + c) * 2 + 1];
        }
        float mean = s * invN;
        float var  = s2 * invN - mean * mean;   // E[x^2]-E[x]^2, matches ref
        float sc   = gamma[t] * rsqrtf(var + 1e-5f);
        scalebias[2 * t + 0] = sc;
        scalebias[2 * t + 1] = beta[t] - mean * sc;
    }
}

// ---------------------------------------------------------------------------
// Kernel 3: fused BN-affine + ReLU + gate projection via WMMA fp32.
//   gx[g][t][b] = relu(bn(x[b,t,:])) . w_ih[g,:] + b_ih[g]
// Each wave handles 16 consecutive flattened rows r = b*T + t.
// A layout (16x4 f32, ISA 7.12.2): lane L<16 -> row L, K={k0,k0+1};
//                                  lane L>=16 -> row L-16, K={k0+2,k0+3}.
// B layout (4x16): lane L<16 -> N=L, K={k0,k0+1}; lane>=16 -> N=L-16, K+2.
// D layout: VGPR v, lanes 0-15 -> M=v, N=lane; lanes 16-31 -> M=v+8, N=lane-16.
//
// B fragments are loop-invariant: preloaded once via a CLAMPED in-bounds
// address + 0/1 mask multiply (no divergent loads).  t/b indices derived
// incrementally from the uniform rowBase (no per-lane 64-bit div chains).
// ---------------------------------------------------------------------------
__global__ void gx_wmma_kernel(const float* __restrict__ x,
                               const float* __restrict__ w_ih,   // [3][24]
                               const float* __restrict__ b_ih,   // [3]
                               const float* __restrict__ scalebias, // [T][2]
                               float* __restrict__ gx,           // 3 x [T*Bn]
                               int Bn, int totalRows) {
    const int lane   = threadIdx.x & 31;
    const int wave   = threadIdx.x >> 5;
    const int mlocal = lane & 15;
    const int hi     = lane >> 4;                     // 0: K 0,1 | 1: K 2,3
    const int rowBase = (blockIdx.x * (blockDim.x >> 5) + wave) * 16;

    // Uniform tile origin in (b, t) coordinates (scalar div on uniform value).
    const int bBase = rowBase / T_DIM;
    const int tBase = rowBase - bBase * T_DIM;

    int r     = rowBase + mlocal;
    int rload = (r < totalRows) ? r : (totalRows - 1); // clamp, keep EXEC full
    int tl    = tBase + mlocal;                        // row's t (<=1 wrap)
    if (tl >= T_DIM) tl -= T_DIM;
    float sc  = scalebias[2 * tl + 0];
    float bs  = scalebias[2 * tl + 1];
    const float* xrow = x + rload * F_DIM + hi * 2;

    const int   n    = mlocal;            // gate column for this lane's B frag
    const bool  hasB = (n < 3);
    const int   nc   = hasB ? n : 0;      // clamped -> always in-bounds load
    const float msk  = hasB ? 1.0f : 0.0f;

    // Preload all 6 loop-invariant B fragments (unconditional b64 loads).
    v2f bfrag[6];
#pragma unroll
    for (int j = 0; j < 6; ++j) {
        const int kk = j * 4 + hi * 2;
        float2 wv = *(const float2*)(w_ih + nc * F_DIM + kk);  // 8B aligned
        bfrag[j].x = wv.x * msk;   // exact zero for unused N columns
        bfrag[j].y = wv.y * msk;
    }

    v8f c = {};
#pragma unroll
    for (int j = 0; j < 6; ++j) {
        float2 xv = *(const float2*)(xrow + j * 4);            // 8B aligned
        v2f a;
        a.x = fmaxf(fmaf(xv.x, sc, bs), 0.f);
        a.y = fmaxf(fmaf(xv.y, sc, bs), 0.f);
        // D = A(16x4) * B(4x16) + C   (fp32 exact, matches reference math)
        c = __builtin_amdgcn_wmma_f32_16x16x4_f32(
                /*neg_a=*/false, a, /*neg_b=*/false, bfrag[j],
                /*c_mod=*/(short)0, c, /*reuse_a=*/false, /*reuse_b=*/false);
    }

    // Lanes {0,1,2} own gates for rows M=0..7; lanes {16,17,18} for M=8..15.
    if (hasB) {
        const float bi = b_ih[n];
        float* plane = gx + n * (T_DIM * Bn);
        const int o0 = hi * 8;
        int tt = tBase + o0;                 // <= 79 + 8 -> at most one wrap
        int bb = bBase;
        if (tt >= T_DIM) { tt -= T_DIM; ++bb; }
#pragma unroll
        for (int v = 0; v < 8; ++v) {
            int rr = rowBase + o0 + v;
            if (rr < totalRows) plane[tt * Bn + bb] = c[v] + bi;
            if (++tt >= T_DIM) { tt = 0; ++bb; }
        }
    }
}

// ---------------------------------------------------------------------------
// Kernel 4: per-sample GRU(h=1) recurrence over T + fused ReLU + FC dot.
// gx planes are [t][b] so loads across the wave are fully coalesced.
// ---------------------------------------------------------------------------
__global__ void gru_fc_kernel(const float* __restrict__ gx,
                              const float* __restrict__ w_hh, // [3]
                              const float* __restrict__ b_hh, // [3]
                              const float* __restrict__ fc_w, // [T]
                              const float* __restrict__ fc_b, // [1]
                              float* __restrict__ out, int Bn) {
    int b = blockIdx.x * blockDim.x + threadIdx.x;
    if (b >= Bn) return;
    const float wr = w_hh[0], wz = w_hh[1], wn = w_hh[2];
    const float br = b_hh[0], bz = b_hh[1], bn = b_hh[2];
    const float* gr = gx;
    const float* gz = gx + T_DIM * Bn;
    const float* gn = gx + 2 * T_DIM * Bn;

    float h = 0.f, acc = 0.f;
    for (int t = 0; t < T_DIM; ++t) {
        int idx = t * Bn + b;
        float rv = 1.f / (1.f + expf(-(gr[idx] + h * wr + br)));
        float zv = 1.f / (1.f + expf(-(gz[idx] + h * wz + bz)));
        float nv = tanhf(gn[idx] + rv * (h * wn + bn));
        h = (1.f - zv) * nv + zv * h;
        acc = fmaf(fc_w[t], fmaxf(h, 0.f), acc);
    }
    out[b] = acc + fc_b[0];
}

// ---------------------------------------------------------------------------
extern "C" void kernel_launch(void* const* d_in, const int* in_sizes, int n_in,
                              void* d_out, int out_size, void* d_ws, size_t ws_size,
                              hipStream_t stream) {
    const float* x      = (const float*)d_in[0];
    const float* gamma  = (const float*)d_in[1];
    const float* beta   = (const float*)d_in[2];
    const float* w_ih   = (const float*)d_in[3];
    const float* w_hh   = (const float*)d_in[4];
    const float* b_ih   = (const float*)d_in[5];
    const float* b_hh   = (const float*)d_in[6];
    const float* fc_w   = (const float*)d_in[7];
    const float* fc_b   = (const float*)d_in[8];
    float*       out    = (float*)d_out;

    const int Bn        = in_sizes[0] / (T_DIM * F_DIM);   // 16384
    const int totalRows = Bn * T_DIM;                      // 1,310,720

    // Workspace layout (floats):
    //   [0, 3*T*Bn)                  gx planes (r,z,n)
    //   [+, T*nchunk*2)              stats partials
    //   [+, T*2)                     per-t (scale, bias)
    const int bPerChunk = 1024;
    const int nchunk    = (Bn + bPerChunk - 1) / bPerChunk;   // 16

    float* ws        = (float*)d_ws;
    float* gx        = ws;
    float* part      = ws + 3l * T_DIM * Bn;
    float* scalebias = part + (long)T_DIM * nchunk * 2;

    // 1) per-t partial sums over (B,F)
    dim3 gs1(T_DIM, nchunk);
    stats_partial_kernel<<<gs1, 256, 0, stream>>>(x, part, Bn, nchunk, bPerChunk);

    // 2) finalize -> per-t affine scale/bias
    stats_final_kernel<<<1, 128, 0, stream>>>(part, gamma, beta, scalebias,
                                              nchunk, 1.f / ((float)Bn * F_DIM));

    // 3) fused BN + ReLU + WMMA gate projection (8 waves x 16 rows per block)
    const int rowsPerBlock = 128;
    const int nblocksA = (totalRows + rowsPerBlock - 1) / rowsPerBlock;
    gx_wmma_kernel<<<nblocksA, 256, 0, stream>>>(x, w_ih, b_ih, scalebias, gx,
                                                 Bn, totalRows);

    // 4) recurrence + FC
    gru_fc_kernel<<<(Bn + 255) / 256, 256, 0, stream>>>(gx, w_hh, b_hh,
                                                        fc_w, fc_b, out, Bn);
}